// OutlierRNN_34488587387185
// MI455X (gfx1250) — compile-verified
//
#include <hip/hip_runtime.h>
#include <cstdint>

#define DEV __device__ __forceinline__

typedef float v2f  __attribute__((ext_vector_type(2)));
typedef float v8f  __attribute__((ext_vector_type(8)));
typedef unsigned int u32x4 __attribute__((ext_vector_type(4)));
typedef int   i32x4 __attribute__((ext_vector_type(4)));
typedef int   i32x8 __attribute__((ext_vector_type(8)));

#ifndef __has_builtin
#define __has_builtin(x) 0
#endif

#if __has_builtin(__builtin_amdgcn_tensor_load_to_lds) && __has_builtin(__builtin_amdgcn_s_wait_tensorcnt)
#define USE_TDM 1
#else
#define USE_TDM 0
#endif

constexpr int Bsz  = 4096;
constexpr int Tsz  = 2048;
constexpr int DIN  = 8;
constexpr int Fsz  = 4;
constexpr int TC   = 8;              // timesteps staged per chunk
constexpr int NC   = Tsz / TC;       // 256 chunks
constexpr int ROWS = 64;             // batch rows per workgroup (4 waves x 16)
constexpr int RSTR = TC * DIN + 4;   // 68-dword padded LDS row stride (bank-conflict free)
constexpr int BUFSZ = ROWS * RSTR;   // 4352 dwords per buffer

// D = A(16x4,f32) x B(4x16,f32) + C(16x16,f32)
DEV v8f wmma4(v2f a, v2f b, v8f c) {
  return __builtin_amdgcn_wmma_f32_16x16x4_f32(false, a, false, b, (short)0, c, false, false);
}

DEV float fast_tanh(float v) {
#if __has_builtin(__builtin_amdgcn_tanhf)
  return __builtin_amdgcn_tanhf(v);   // v_tanh_f32
#else
  return tanhf(v);
#endif
}

#if USE_TDM
// 2D TDM: tile = (TC*DIN) elems x ROWS rows of f32, row stride T*DIN elems,
// LDS padding: 4 dwords after every 64 dwords -> 68-dword row stride in LDS.
DEV void tdm_load_chunk(const float* xg, int row0, int t0, const float* ldsDst) {
  uint64_t ga = (uint64_t)(uintptr_t)(xg + ((size_t)row0 * Tsz + t0) * DIN);
  uint32_t ldsOff = (uint32_t)(uintptr_t)ldsDst;
  u32x4 g0;
  g0[0] = 1u;                                            // count=1, user descriptor
  g0[1] = ldsOff;                                        // lds_addr (bytes)
  g0[2] = (uint32_t)ga;                                  // global_addr[31:0]
  g0[3] = (uint32_t)((ga >> 32) & 0x01FFFFFFull) | 0x80000000u; // addr[56:32] | type=2
  i32x8 g1;
  g1[0] = (int)((2u << 16)      // data_size = 4B
              | (1u << 20)      // pad_enable
              | (5u << 22)      // pad_interval: 2^(5+1)=64 dwords
              | (3u << 25));    // pad_amount: 3+1 = 4 dwords
  g1[1] = (int)((uint32_t)(Tsz * DIN) << 16);                     // tensor_dim0[15:0] @ [63:48]
  g1[2] = (int)(((uint32_t)(Tsz * DIN) >> 16) | ((uint32_t)Bsz << 16)); // dim0 hi | dim1 lo
  g1[3] = (int)(((uint32_t)Bsz >> 16) | ((uint32_t)(TC * DIN) << 16));  // dim1 hi | tile_dim0
  g1[4] = (int)(uint32_t)ROWS;                                    // tile_dim1 (tile_dim2=0)
  g1[5] = (int)(uint32_t)(Tsz * DIN);                             // tensor_dim0_stride lo32
  g1[6] = 0;
  g1[7] = 0;
  i32x4 gz = {0, 0, 0, 0};
#if defined(__clang_major__) && __clang_major__ >= 23
  i32x8 gz8 = {0, 0, 0, 0, 0, 0, 0, 0};
  __builtin_amdgcn_tensor_load_to_lds(g0, g1, gz, gz, gz8, 0);
#else
  __builtin_amdgcn_tensor_load_to_lds(g0, g1, gz, gz, 0);
#endif
}
#endif

__global__ __launch_bounds__(128) void rnn_fused(
    const float* __restrict__ x,  const float* __restrict__ W0, const float* __restrict__ b0,
    const float* __restrict__ Wi, const float* __restrict__ bi, const float* __restrict__ Wh,
    const float* __restrict__ W1, const float* __restrict__ b1,
    const float* __restrict__ W2, const float* __restrict__ b2,
    float* __restrict__ out)
{
  __shared__ float xs[2][BUFSZ];   // double-buffered staged x tiles (~34 KB)

  const int tid  = threadIdx.x;
  const int wave = tid >> 5;
  const int lane = tid & 31;
  const int half = lane >> 4;      // 0: K=0,1   1: K=2,3 (A/B operand halves)
  const int col  = lane & 15;      // M for A-layout, N (batch row) for B/D-layout
  const int row0 = blockIdx.x * ROWS;
  const int ldsRow = wave * 16 + col;
  const int kb   = 2 * half;
  const bool mok = (col < Fsz);    // valid output-feature rows of the transposed weights

  // A-operands: transposed weights, zero-padded to 16 rows. A[M][K] = W[K][M].
  v2f w0lo, w0hi, wiT, whT;
  w0lo[0] = mok ? W0[(kb + 0) * Fsz + col] : 0.f;
  w0lo[1] = mok ? W0[(kb + 1) * Fsz + col] : 0.f;
  w0hi[0] = mok ? W0[(kb + 4) * Fsz + col] : 0.f;
  w0hi[1] = mok ? W0[(kb + 5) * Fsz + col] : 0.f;
  wiT[0]  = mok ? Wi[(kb + 0) * Fsz + col] : 0.f;
  wiT[1]  = mok ? Wi[(kb + 1) * Fsz + col] : 0.f;
  whT[0]  = mok ? Wh[(kb + 0) * Fsz + col] : 0.f;
  whT[1]  = mok ? Wh[(kb + 1) * Fsz + col] : 0.f;

  // Bias preloaded into the C operand (C[M=r][N] = b[r], rows 0..3, lanes 0..15).
  v8f bias0, biasI;
#pragma unroll
  for (int r = 0; r < 8; ++r) {
    bias0[r] = (half == 0 && r < Fsz) ? b0[r] : 0.f;
    biasI[r] = (half == 0 && r < Fsz) ? bi[r] : 0.f;
  }

  v2f cB = {0.f, 0.f};   // carry c^T held in WMMA B-operand layout
  int p = 0;

#if USE_TDM
  if (wave == 0) {
    tdm_load_chunk(x, row0, 0, &xs[0][0]);
    __builtin_amdgcn_s_wait_tensorcnt(0);
  }
  __syncthreads();
#endif

  for (int c = 0; c < NC; ++c) {
#if USE_TDM
    if (wave == 0 && (c + 1) < NC)
      tdm_load_chunk(x, row0, (c + 1) * TC, &xs[p ^ 1][0]);   // async prefetch next chunk
#else
    {
      const int t0 = c * TC;
      for (int s = tid; s < ROWS * (TC * DIN / 4); s += 128) {
        const int r = s >> 4, q = s & 15;
        const float4 v = *(const float4*)(x + ((size_t)(row0 + r) * Tsz + t0) * DIN + q * 4);
        *(float4*)&xs[0][r * RSTR + q * 4] = v;
      }
      __syncthreads();
    }
#endif

    const float* xb = &xs[p][ldsRow * RSTR];
#pragma unroll
    for (int t = 0; t < TC; ++t) {
      // x_t^T as B-operand: lane(half,col) holds x[row=col][k = 2*half + i]
      const float* rp = xb + t * DIN + kb;
      v2f xlo = { rp[0], rp[1] };
      v2f xhi = { rp[4], rp[5] };

      // h^T = relu(W0^T x_t^T + b0) : two chained K=4 WMMAs (off the carry chain)
      v8f h = wmma4(w0lo, xlo, bias0);
      h = wmma4(w0hi, xhi, h);
      float h0 = fmaxf(h[0], 0.f), h1 = fmaxf(h[1], 0.f);
      float h2 = fmaxf(h[2], 0.f), h3 = fmaxf(h[3], 0.f);
      // D-layout -> B-layout: rows 2,3 (lanes 0-15) move to lanes 16-31 of vgpr 0,1
      float s2 = __shfl(h2, col, 32);
      float s3 = __shfl(h3, col, 32);
      v2f hB = { half ? s2 : h0, half ? s3 : h1 };

      // c'^T = tanh(Wi^T h^T + Wh^T c^T + bi) : two chained WMMAs + v_tanh
      v8f g = wmma4(wiT, hB, biasI);
      v8f d = wmma4(whT, cB, g);
      float c0 = fast_tanh(d[0]), c1 = fast_tanh(d[1]);
      float c2 = fast_tanh(d[2]), c3 = fast_tanh(d[3]);
      float u2 = __shfl(c2, col, 32);
      float u3 = __shfl(c3, col, 32);
      cB[0] = half ? u2 : c0;
      cB[1] = half ? u3 : c1;
    }

#if USE_TDM
    if (wave == 0) __builtin_amdgcn_s_wait_tensorcnt(0);  // prefetched chunk landed
    __syncthreads();
    p ^= 1;
#else
    __syncthreads();
#endif
  }

  // Head on the final carry: y^T = W2^T relu(W1^T c^T + b1) + b2
  v2f w1T, w2T;
  w1T[0] = mok ? W1[(kb + 0) * Fsz + col] : 0.f;
  w1T[1] = mok ? W1[(kb + 1) * Fsz + col] : 0.f;
  w2T[0] = mok ? W2[(kb + 0) * Fsz + col] : 0.f;
  w2T[1] = mok ? W2[(kb + 1) * Fsz + col] : 0.f;
  v8f bC1, bC2;
#pragma unroll
  for (int r = 0; r < 8; ++r) {
    bC1[r] = (half == 0 && r < Fsz) ? b1[r] : 0.f;
    bC2[r] = (half == 0 && r < Fsz) ? b2[r] : 0.f;
  }
  v8f hd = wmma4(w1T, cB, bC1);
  float q0 = fmaxf(hd[0], 0.f), q1 = fmaxf(hd[1], 0.f);
  float q2 = fmaxf(hd[2], 0.f), q3 = fmaxf(hd[3], 0.f);
  float r2 = __shfl(q2, col, 32);
  float r3 = __shfl(q3, col, 32);
  v2f hB2 = { half ? r2 : q0, half ? r3 : q1 };
  v8f yd = wmma4(w2T, hB2, bC2);

  if (half == 0) {
    float* o = out + (size_t)(row0 + ldsRow) * Fsz;
    o[0] = yd[0]; o[1] = yd[1]; o[2] = yd[2]; o[3] = yd[3];
  }
}

extern "C" void kernel_launch(void* const* d_in, const int* in_sizes, int n_in,
                              void* d_out, int out_size, void* d_ws, size_t ws_size,
                              hipStream_t stream) {
  (void)in_sizes; (void)n_in; (void)out_size; (void)d_ws; (void)ws_size;
  rnn_fused<<<Bsz / ROWS, 128, 0, stream>>>(
      (const float*)d_in[0], (const float*)d_in[1], (const float*)d_in[2],
      (const float*)d_in[3], (const float*)d_in[4], (const float*)d_in[5],
      (const float*)d_in[6], (const float*)d_in[7], (const float*)d_in[8],
      (const float*)d_in[9], (float*)d_out);
}